// FFTConv_48335561949935
// MI455X (gfx1250) — compile-verified
//
#include <hip/hip_runtime.h>
#include <math.h>

// FFT-based long convolution for MI455X (gfx1250):
//   y[b,d,:] = linearconv(h[d,:], x[b,d,:])[:L] + x[b,d,:]*B[d]
// One workgroup per (b,d). Entire 16384-pt complex FFT lives in LDS
// (2 x 128KB ping-pong <= 320KB/WGP). Radix-16 butterflies are complex
// 16x16x16 matmuls against the DFT16 matrix done with chained
// v_wmma_f32_16x16x4_f32 (4 real matmuls x 4 K-chunks = 16 WMMA / tile).
// fp32 WMMA has no A/B negate (NEG = {CNeg,0,0}), so we keep a negated
// copy of the imaginary DFT matrix for the Dr -= Wi*Bi term.
// Stage twiddles are generated incrementally (w, w^2, w^4, w^8, then one
// complex mul per u-step) -> 2 TRANS ops per tile instead of 16.

typedef __attribute__((ext_vector_type(2))) float v2f;
typedef __attribute__((ext_vector_type(8))) float v8f;

#define NFFT   16384
#define LSIG   8192
#define COLS   (NFFT / 16)   // 1024 columns per radix-16 stage
#define DMODEL 1024
#define TWOPI_OVER_N 3.834951969714103e-4f   // 2*pi/16384

// ---------------- radix-16 Stockham stage (WMMA) ----------------
// in/out: complex float2 arrays of NFFT elements in LDS.
// s: current stride (1, 16, 256). Columns c in [0, COLS):
//   gather:  in[c + COLS*t],  t = 0..15   (B matrix, rows=t, cols=c)
//   D = W16 * B  (complex),  W16[u][t] = exp(-2*pi*i*u*t/16)  (A matrix)
//   twiddle: * exp(-2*pi*i * p*u*s / NFFT),  p = c / s
//   scatter: out[(c%s) + 16*s*p + s*u]
__device__ __forceinline__ void radix16_stage(const float2* __restrict__ in,
                                              float2* __restrict__ out,
                                              int s, int tid,
                                              const v2f* wr, const v2f* wi,
                                              const v2f* wni) {
  const int lane = tid & 31;
  const int wave = tid >> 5;
  const int colL = lane & 15;   // matrix column within tile (N index)
  const int half = lane >> 4;   // lane half selects K rows 0..1 vs 2..3

  for (int tile = wave; tile < COLS / 16; tile += 8) {
    const int c = tile * 16 + colL;

    // Gather B matrix (complex) in WMMA B layout:
    // lane l, vgpr v -> K = 4*kk + 2*(l>=16) + v, N = l&15
    v2f br[4], bi[4];
#pragma unroll
    for (int kk = 0; kk < 4; ++kk) {
#pragma unroll
      for (int v = 0; v < 2; ++v) {
        const int t = kk * 4 + half * 2 + v;
        const float2 z = in[c + COLS * t];
        br[kk][v] = z.x;
        bi[kk][v] = z.y;
      }
    }

    // Complex matmul: Dr = Wr*Br + (-Wi)*Bi ; Di = Wr*Bi + Wi*Br
    v8f dr = {0.f, 0.f, 0.f, 0.f, 0.f, 0.f, 0.f, 0.f};
    v8f di = {0.f, 0.f, 0.f, 0.f, 0.f, 0.f, 0.f, 0.f};
#pragma unroll
    for (int kk = 0; kk < 4; ++kk) {
      dr = __builtin_amdgcn_wmma_f32_16x16x4_f32(false, wr[kk],  false, br[kk],
                                                 (short)0, dr, false, false);
      dr = __builtin_amdgcn_wmma_f32_16x16x4_f32(false, wni[kk], false, bi[kk],
                                                 (short)0, dr, false, false);
      di = __builtin_amdgcn_wmma_f32_16x16x4_f32(false, wr[kk],  false, bi[kk],
                                                 (short)0, di, false, false);
      di = __builtin_amdgcn_wmma_f32_16x16x4_f32(false, wi[kk],  false, br[kk],
                                                 (short)0, di, false, false);
    }

    // Twiddle + scatter. D layout: vgpr v, lane -> (M = v + 8*half, N = colL)
    // tw(u) = w^u, w = exp(-2*pi*i*p*s/N). Build w^8 by squaring, then one
    // complex mul per u-step: 1 sincos per tile instead of 8.
    const int p = c / s;          // s is a power of two (compile-time literal)
    const int q = c - p * s;
    const int obase = q + 16 * s * p;

    const float phi = -TWOPI_OVER_N * (float)(p * s);
    float w1i, w1r;
    __sincosf(phi, &w1i, &w1r);                       // w^1
    const float w2r = w1r * w1r - w1i * w1i;          // w^2
    const float w2i = 2.f * w1r * w1i;
    const float w4r = w2r * w2r - w2i * w2i;          // w^4
    const float w4i = 2.f * w2r * w2i;
    const float w8r = w4r * w4r - w4i * w4i;          // w^8
    const float w8i = 2.f * w4r * w4i;
    float twr = half ? w8r : 1.f;                     // start at u = 8*half
    float twi = half ? w8i : 0.f;

#pragma unroll
    for (int v = 0; v < 8; ++v) {
      const float yr = dr[v] * twr - di[v] * twi;
      const float yi = dr[v] * twi + di[v] * twr;
      out[obase + s * (v + 8 * half)] = make_float2(yr, yi);
      const float ntr = twr * w1r - twi * w1i;        // tw *= w
      const float nti = twr * w1i + twi * w1r;
      twr = ntr;
      twi = nti;
    }
  }
}

// ---------------- final radix-4 stage (s = 4096, twiddles == 1) -------------
__device__ __forceinline__ void radix4_stage(const float2* __restrict__ in,
                                             float2* __restrict__ out, int tid) {
  for (int c = tid; c < 4096; c += 256) {
    const float2 a0 = in[c];
    const float2 a1 = in[c + 4096];
    const float2 a2 = in[c + 8192];
    const float2 a3 = in[c + 12288];
    const float s02r = a0.x + a2.x, s02i = a0.y + a2.y;
    const float d02r = a0.x - a2.x, d02i = a0.y - a2.y;
    const float s13r = a1.x + a3.x, s13i = a1.y + a3.y;
    const float d13r = a1.x - a3.x, d13i = a1.y - a3.y;
    out[c]         = make_float2(s02r + s13r, s02i + s13i);  // X0
    out[c + 4096]  = make_float2(d02r + d13i, d02i - d13r);  // X1 = d02 - i*d13
    out[c + 8192]  = make_float2(s02r - s13r, s02i - s13i);  // X2
    out[c + 12288] = make_float2(d02r - d13i, d02i + d13r);  // X3 = d02 + i*d13
  }
}

// ---------------- main kernel ----------------
extern "C" __global__ void __launch_bounds__(256, 1)
fftconv_gfx1250(const float* __restrict__ h, const float* __restrict__ x,
                const float* __restrict__ Bg, float* __restrict__ out) {
  extern __shared__ char smem_raw[];
  float2* buf0 = (float2*)smem_raw;       // 128 KB
  float2* buf1 = buf0 + NFFT;             // 128 KB

  const int tid = threadIdx.x;
  const int blk = blockIdx.x;
  const int d = blk & (DMODEL - 1);
  const int b = blk >> 10;

  const float* hrow = h + (size_t)d * LSIG;
  const float* xrow = x + ((size_t)b * DMODEL + d) * LSIG;

  // Per-lane DFT16 matrix chunks in WMMA A layout:
  // lane l, vgpr v -> K = 4*kk + 2*(l>=16) + v, M = l&15
  const int lane = tid & 31;
  const int ua = lane & 15;
  const int halfa = lane >> 4;
  v2f wr[4], wi[4], wni[4];
#pragma unroll
  for (int kk = 0; kk < 4; ++kk) {
#pragma unroll
    for (int v = 0; v < 2; ++v) {
      const int t = kk * 4 + halfa * 2 + v;
      const float ang = -0.39269908169872414f * (float)(ua * t);  // -2pi/16
      float sn, cs;
      __sincosf(ang, &sn, &cs);
      wr[kk][v] = cs;
      wi[kk][v] = sn;
      wni[kk][v] = -sn;
    }
  }

  // Pack z = h + i*x, zero-pad to 2L (linear, non-circular convolution)
  for (int j = tid; j < LSIG; j += 256)
    buf0[j] = make_float2(hrow[j], xrow[j]);
  for (int j = LSIG + tid; j < NFFT; j += 256)
    buf0[j] = make_float2(0.f, 0.f);
  __syncthreads();

  // Forward 16384-pt complex FFT: radices 16,16,16,4 (Stockham ping-pong)
  radix16_stage(buf0, buf1, 1,   tid, wr, wi, wni); __syncthreads();
  radix16_stage(buf1, buf0, 16,  tid, wr, wi, wni); __syncthreads();
  radix16_stage(buf0, buf1, 256, tid, wr, wi, wni); __syncthreads();
  radix4_stage (buf1, buf0, tid);                   __syncthreads();

  // Unpack H,X from packed spectrum Z; Y = H*X; stage conj(Y) for inverse.
  // H[k] = (Z[k] + conj(Z[N-k]))/2 ; X[k] = (Z[k] - conj(Z[N-k]))/(2i)
  for (int k = tid; k < NFFT; k += 256) {
    const int kr = (NFFT - k) & (NFFT - 1);
    const float2 z1 = buf0[k];
    const float2 z2 = buf0[kr];
    const float hr = 0.5f * (z1.x + z2.x);
    const float hi = 0.5f * (z1.y - z2.y);
    const float nr = z1.x - z2.x;
    const float ni = z1.y + z2.y;
    const float xr = 0.5f * ni;
    const float xi = -0.5f * nr;
    const float yr = hr * xr - hi * xi;
    const float yi = hr * xi + hi * xr;
    buf1[k] = make_float2(yr, -yi);  // conj for inverse-via-forward trick
  }
  __syncthreads();

  // Inverse FFT: IFFT(Y) = conj(FFT(conj(Y)))/N ; we only need the real part.
  radix16_stage(buf1, buf0, 1,   tid, wr, wi, wni); __syncthreads();
  radix16_stage(buf0, buf1, 16,  tid, wr, wi, wni); __syncthreads();
  radix16_stage(buf1, buf0, 256, tid, wr, wi, wni); __syncthreads();
  radix4_stage (buf0, buf1, tid);                   __syncthreads();

  // Epilogue: y = Re(.)/N + x*B[d]
  const float bd = Bg[d];
  float* orow = out + ((size_t)b * DMODEL + d) * LSIG;
  const float inv = 1.0f / (float)NFFT;
  for (int j = tid; j < LSIG; j += 256)
    orow[j] = buf1[j].x * inv + xrow[j] * bd;
}

// ---------------- host launcher ----------------
extern "C" void kernel_launch(void* const* d_in, const int* in_sizes, int n_in,
                              void* d_out, int out_size, void* d_ws, size_t ws_size,
                              hipStream_t stream) {
  (void)in_sizes; (void)n_in; (void)out_size; (void)d_ws; (void)ws_size;
  const float* h = (const float*)d_in[0];  // (1024, 8192)
  const float* x = (const float*)d_in[1];  // (2, 1024, 8192)
  const float* B = (const float*)d_in[2];  // (1024, 1)
  float* out = (float*)d_out;              // (2, 1024, 8192)

  const size_t lds_bytes = (size_t)2 * NFFT * sizeof(float2);  // 256 KB
  hipError_t err = hipFuncSetAttribute(
      reinterpret_cast<const void*>(fftconv_gfx1250),
      hipFuncAttributeMaxDynamicSharedMemorySize, (int)lds_bytes);
  (void)err;

  dim3 grid(2 * DMODEL);  // one workgroup per (batch, channel)
  fftconv_gfx1250<<<grid, 256, lds_bytes, stream>>>(h, x, B, out);
}